// GCNModel3_13804024889640
// MI455X (gfx1250) — compile-verified
//
#include <hip/hip_runtime.h>
#include <math.h>

#define NN 50000
#define NE 800000
#define FW 128   // feature width for all SpMM stages

typedef __attribute__((ext_vector_type(2))) float v2f;
typedef __attribute__((ext_vector_type(8))) float v8f;

// ---------------- zero ----------------
__global__ __launch_bounds__(256) void zero_kernel(float4* p, int n4) {
    int i = blockIdx.x * blockDim.x + threadIdx.x;
    if (i < n4) p[i] = make_float4(0.f, 0.f, 0.f, 0.f);
}

// ---------------- degrees (float atomics; exact for integer counts) -------
__global__ __launch_bounds__(256) void degree_kernel(const int* __restrict__ src,
                                                     const int* __restrict__ dst,
                                                     float* outd, float* ind, int E) {
    int e = blockIdx.x * blockDim.x + threadIdx.x;
    if (e < E) {
        atomicAdd(&outd[src[e]], 1.0f);
        atomicAdd(&ind[dst[e]], 1.0f);
    }
}

// ---------------- norms ----------------
__global__ __launch_bounds__(256) void norm_kernel(const float* __restrict__ outd,
                                                   const float* __restrict__ ind,
                                                   float* sn, float* dn, int n) {
    int i = blockIdx.x * blockDim.x + threadIdx.x;
    if (i >= n) return;
    float o = outd[i];
    float d = ind[i];
    sn[i] = (o > 0.f) ? rsqrtf(o) : 0.f;
    dn[i] = (d > 0.f) ? rsqrtf(d) : 0.f;
}

// ---------------- xs = x * src_norm (row-wise), 128-wide rows -------------
__global__ __launch_bounds__(256) void scale_kernel(const float4* __restrict__ x,
                                                    const float* __restrict__ sn,
                                                    float4* __restrict__ xs, int total4) {
    int i = blockIdx.x * blockDim.x + threadIdx.x;
    if (i >= total4) return;
    int node = i >> 5;          // 32 float4 per 128-wide row
    float s = sn[node];
    float4 v = x[i];
    v.x *= s; v.y *= s; v.z *= s; v.w *= s;
    xs[i] = v;
}

// ---------------- SpMM: one wave per edge, float4 per lane ----------------
__global__ __launch_bounds__(256) void spmm_kernel(const float* __restrict__ xs,
                                                   const int* __restrict__ src,
                                                   const int* __restrict__ dst,
                                                   float* __restrict__ agg, int E) {
    int wave = blockIdx.x * (blockDim.x >> 5) + (threadIdx.x >> 5);
    int lane = threadIdx.x & 31;
    if (wave >= E) return;
    int s = src[wave];
    int d = dst[wave];
    float4 v = ((const float4*)(xs + (size_t)s * FW))[lane];
    float* dp = agg + (size_t)d * FW + lane * 4;
    atomicAdd(dp + 0, v.x);
    atomicAdd(dp + 1, v.y);
    atomicAdd(dp + 2, v.z);
    atomicAdd(dp + 3, v.w);
}

// ---- fused agg @ W row-strip GEMM via f32 WMMA, epilogue dn-scale+bias+ReLU
// One wave computes a 16 x N output strip: NT = N/16 independent 16x16
// accumulators, A fragment loaded once per K-step and reused across all NT
// WMMAs (independent accumulation chains pipeline in the matrix unit).
// Layouts per CDNA5 ISA 7.12.2:
//   A 16x4:  lane<16 -> M=lane, K={k0,k0+1}; lane>=16 -> M=lane-16, K={k0+2,k0+3}
//   B 4x16:  vgpr j, lane<16 -> K=k0+j, N=lane; lane>=16 -> K=k0+2+j, N=lane-16
//   C/D:     vgpr v, M = v + (lane<16 ? 0 : 8), N = lane&15
// dn-scaling is hoisted to the epilogue: (diag(dn)*agg)@W = diag(dn)*(agg@W).
template <int K, int N>
__global__ __launch_bounds__(256) void gemm_rowstrip_wmma(const float* __restrict__ agg,
                                                          const float* __restrict__ dn,
                                                          const float* __restrict__ W,
                                                          const float* __restrict__ bias,
                                                          float* __restrict__ out,
                                                          int totalWaves) {
    constexpr int NT = N / 16;
    int wave = blockIdx.x * (blockDim.x >> 5) + (threadIdx.x >> 5);
    if (wave >= totalWaves) return;
    int lane = threadIdx.x & 31;
    int hl   = lane & 15;
    int koff = (lane < 16) ? 0 : 2;
    int rowA = wave * 16 + hl;
    const float* arow = agg + (size_t)rowA * K;

    v8f c[NT] = {};

#pragma unroll 2
    for (int k0 = 0; k0 < K; k0 += 4) {
        v2f a;
        a.x = arow[k0 + koff];
        a.y = arow[k0 + koff + 1];
#pragma unroll
        for (int t = 0; t < NT; ++t) {
            v2f b;
            b.x = W[(size_t)(k0 + koff)     * N + t * 16 + hl];
            b.y = W[(size_t)(k0 + koff + 1) * N + t * 16 + hl];
            c[t] = __builtin_amdgcn_wmma_f32_16x16x4_f32(
                       false, a, false, b, (short)0, c[t], false, false);
        }
    }

    float bv[NT];
#pragma unroll
    for (int t = 0; t < NT; ++t) bv[t] = bias[t * 16 + hl];

    int mbase = wave * 16 + ((lane < 16) ? 0 : 8);
#pragma unroll
    for (int v = 0; v < 8; ++v) {
        float rn = dn[mbase + v];          // broadcast within half-wave
        float* orow = out + (size_t)(mbase + v) * N;
#pragma unroll
        for (int t = 0; t < NT; ++t) {
            float val = c[t][v] * rn + bv[t];
            orow[t * 16 + hl] = val > 0.f ? val : 0.f;
        }
    }
}

// ---------------- final 64->1 projection + sigmoid ------------------------
__global__ __launch_bounds__(256) void proj_kernel(const float* __restrict__ x,
                                                   const float* __restrict__ Wp,
                                                   const float* __restrict__ bp,
                                                   float* __restrict__ out, int n) {
    int i = blockIdx.x * blockDim.x + threadIdx.x;
    if (i >= n) return;
    const float4* xr = (const float4*)(x + (size_t)i * 64);
    const float4* wr = (const float4*)Wp;
    float acc = 0.f;
#pragma unroll
    for (int j = 0; j < 16; ++j) {
        float4 a = xr[j];
        float4 w = wr[j];
        acc += a.x * w.x + a.y * w.y + a.z * w.z + a.w * w.w;
    }
    acc += bp[0];
    out[i] = 1.0f / (1.0f + expf(-acc));
}

extern "C" void kernel_launch(void* const* d_in, const int* in_sizes, int n_in,
                              void* d_out, int out_size, void* d_ws, size_t ws_size,
                              hipStream_t stream) {
    const float* features = (const float*)d_in[0];
    const int*   src      = (const int*)d_in[1];
    const int*   dst      = (const int*)d_in[2];
    // d_in[3] = edge_types: unused by the reference forward
    const float* W1 = (const float*)d_in[4];
    const float* b1 = (const float*)d_in[5];
    const float* W2 = (const float*)d_in[6];
    const float* b2 = (const float*)d_in[7];
    const float* W3 = (const float*)d_in[8];
    const float* b3 = (const float*)d_in[9];
    const float* Wp = (const float*)d_in[10];
    const float* bp = (const float*)d_in[11];
    float* out = (float*)d_out;

    float* ws   = (float*)d_ws;
    float* outd = ws;                    // NN
    float* ind  = outd + NN;             // NN
    float* sn   = ind  + NN;             // NN
    float* dn   = sn   + NN;             // NN
    float* B0   = dn   + NN;             // NN*128 : xs scratch
    float* B1   = B0 + (size_t)NN * FW;  // NN*128 : agg scratch
    float* B2   = B1 + (size_t)NN * FW;  // NN*128 : layer output

    auto cdiv = [](int a, int b) { return (a + b - 1) / b; };
    const int TPB = 256;

    // degrees + norms (zero counters every call: deterministic under replay)
    zero_kernel<<<cdiv(2 * NN / 4, TPB), TPB, 0, stream>>>((float4*)outd, 2 * NN / 4);
    degree_kernel<<<cdiv(NE, TPB), TPB, 0, stream>>>(src, dst, outd, ind, NE);
    norm_kernel<<<cdiv(NN, TPB), TPB, 0, stream>>>(outd, ind, sn, dn, NN);

    const int row4    = NN * (FW / 4);       // float4 count of an NN x 128 buffer
    const int spmmBl  = cdiv(NE, TPB / 32);  // one wave per edge
    const int mWaves  = NN / 16;             // 3125 row-strip waves
    const int gemmBl  = cdiv(mWaves, TPB / 32);

    auto layer_front = [&](const float* xin) {
        scale_kernel<<<cdiv(row4, TPB), TPB, 0, stream>>>((const float4*)xin, sn,
                                                          (float4*)B0, row4);
        zero_kernel<<<cdiv(row4, TPB), TPB, 0, stream>>>((float4*)B1, row4);
        spmm_kernel<<<spmmBl, TPB, 0, stream>>>(B0, src, dst, B1, NE);
    };

    // Layer 1: 128 -> 128
    layer_front(features);
    gemm_rowstrip_wmma<128, 128><<<gemmBl, TPB, 0, stream>>>(B1, dn, W1, b1, B2, mWaves);
    // Layer 2: 128 -> 128   (B2 consumed by scale before gemm rewrites it)
    layer_front(B2);
    gemm_rowstrip_wmma<128, 128><<<gemmBl, TPB, 0, stream>>>(B1, dn, W2, b2, B2, mWaves);
    // Layer 3: 128 -> 64
    layer_front(B2);
    gemm_rowstrip_wmma<128, 64><<<gemmBl, TPB, 0, stream>>>(B1, dn, W3, b3, B2, mWaves);

    proj_kernel<<<cdiv(NN, TPB), TPB, 0, stream>>>(B2, Wp, bp, out, NN);
}